// PDA_GNN_50354196578970
// MI455X (gfx1250) — compile-verified
//
#include <hip/hip_runtime.h>
#include <math.h>

#define NUM_NODES 50000
#define DIM 64
#define TILES (NUM_NODES / 16)   // 3125 exactly

typedef __attribute__((ext_vector_type(2))) float v2f;
typedef __attribute__((ext_vector_type(8))) float v8f;

// ---------------- zero fill (float4) ----------------
__global__ void k_zero(float* __restrict__ p, int n4) {
    int i = blockIdx.x * blockDim.x + threadIdx.x;
    if (i < n4) ((float4*)p)[i] = make_float4(0.f, 0.f, 0.f, 0.f);
}

// ---------------- softmax(alpha) -> clip -> renorm ----------------
__global__ void k_weights(const float* __restrict__ alpha, float* __restrict__ w) {
    if (blockIdx.x == 0 && threadIdx.x == 0) {
        float a0 = alpha[0], a1 = alpha[1], a2 = alpha[2];
        float mx = fmaxf(a0, fmaxf(a1, a2));
        float e0 = __expf(a0 - mx), e1 = __expf(a1 - mx), e2 = __expf(a2 - mx);
        float s  = e0 + e1 + e2;
        float w0 = fmaxf(e0 / s, 1e-4f);
        float w1 = fmaxf(e1 / s, 1e-4f);
        float w2 = fmaxf(e2 / s, 1e-4f);
        float inv = 1.0f / (w0 + w1 + w2);
        w[0] = w0 * inv; w[1] = w1 * inv; w[2] = w2 * inv;
    }
}

// ---------------- degree at dst ----------------
__global__ void k_deg(const int* __restrict__ ei, float* __restrict__ deg, int E) {
    int e = blockIdx.x * blockDim.x + threadIdx.x;
    if (e < E) atomicAdd(&deg[ei[E + e]], 1.0f);
}

// ---------------- deg -> d^-1/2 (in place) ----------------
__global__ void k_dinv(float* __restrict__ d, int n) {
    int i = blockIdx.x * blockDim.x + threadIdx.x;
    if (i < n) {
        float dg = d[i];
        d[i] = (dg > 0.f) ? (1.0f / sqrtf(dg)) : 0.f;
    }
}

// ---------------- LGConv: out[dst] += dinv[src]*dinv[dst] * x[src] ----------------
// 16 lanes per edge; each lane moves one float4 of the 64-float row.
// x (12.8MB) is L2-resident on MI455X (192MB L2), so gathers+atomics stay in L2.
__global__ __launch_bounds__(256) void k_lgconv(const int* __restrict__ ei,
                                                const float* __restrict__ dinv,
                                                const float* __restrict__ xin,
                                                float* __restrict__ xout, int E) {
    int gid  = blockIdx.x * blockDim.x + threadIdx.x;
    int edge = gid >> 4;
    int l    = gid & 15;
    if (edge >= E) return;
    if (l == 0) {
        // stream-ahead prefetch of the edge list (global_prefetch_b8)
        __builtin_prefetch(&ei[edge + 4096], 0, 1);
        __builtin_prefetch(&ei[E + edge + 4096], 0, 1);
    }
    int s = ei[edge];
    int d = ei[E + edge];
    float nrm = dinv[s] * dinv[d];
    const float4 v = *(const float4*)(xin + (size_t)s * DIM + l * 4);
    float* o = xout + (size_t)d * DIM + l * 4;
    atomicAdd(o + 0, nrm * v.x);
    atomicAdd(o + 1, nrm * v.y);
    atomicAdd(o + 2, nrm * v.z);
    atomicAdd(o + 3, nrm * v.w);
}

// ---------------- L2-normalize rows via WMMA Gram-diagonal ----------------
// One wave = 16-node tile. A-layout f32 16x4 tile passed as BOTH A and B of
// V_WMMA_F32_16X16X4_F32 => D = X * X^T (Gram); diag = squared row norms.
// widx < 0 : out = normalize(in)           (overwrite)
// widx >= 0: out += w3[widx] * normalize(in)
__global__ __launch_bounds__(256) void k_norm(const float* __restrict__ in,
                                              float* __restrict__ out,
                                              const float* __restrict__ w3,
                                              int widx) {
    int wave = (blockIdx.x * blockDim.x + threadIdx.x) >> 5;
    int lane = threadIdx.x & 31;
    if (wave >= TILES) return;           // wave-uniform: EXEC stays all-ones for WMMA

    int node0 = wave << 4;
    int m     = lane & 15;               // row within tile (A-layout M)
    int khi   = (lane >> 4) << 1;        // K sub-offset: 0 for lanes 0-15, 2 for 16-31

    const float* arow = in + (size_t)(node0 + m) * DIM + khi;
    v8f c = {};
#pragma unroll
    for (int kc = 0; kc < 16; ++kc) {    // K = 64 in chunks of 4
        v2f a = *(const v2f*)(arow + kc * 4);
        c = __builtin_amdgcn_wmma_f32_16x16x4_f32(
                /*neg_a=*/false, a, /*neg_b=*/false, a,
                /*c_mod=*/(short)0, c, /*reuse_a=*/false, /*reuse_b=*/false);
    }

    // Diagonal element for row m lives in VGPR (m&7) at lane (m<8 ? m : m+16).
    int r = m & 7;
    float cr = c[0];
    if (r == 1) cr = c[1];
    if (r == 2) cr = c[2];
    if (r == 3) cr = c[3];
    if (r == 4) cr = c[4];
    if (r == 5) cr = c[5];
    if (r == 6) cr = c[6];
    if (r == 7) cr = c[7];
    int srcLane = (m < 8) ? m : (m + 16);
    float n2  = __shfl(cr, srcLane, 32);
    float inv = 1.0f / fmaxf(sqrtf(n2), 1e-12f);

    float w = (widx >= 0) ? w3[widx] : 1.0f;
    float scale = inv * w;

    // Each lane writes one 32-float half-row (8 x float4).
    size_t roff = (size_t)(node0 + m) * DIM + ((lane >> 4) << 5);
    const float4* srow = (const float4*)(in + roff);
    float4* drow = (float4*)(out + roff);
#pragma unroll
    for (int i = 0; i < 8; ++i) {
        float4 v = srow[i];
        float4 o;
        o.x = v.x * scale; o.y = v.y * scale;
        o.z = v.z * scale; o.w = v.w * scale;
        if (widx >= 0) {
            float4 p = drow[i];
            o.x += p.x; o.y += p.y; o.z += p.z; o.w += p.w;
        }
        drow[i] = o;
    }
}

extern "C" void kernel_launch(void* const* d_in, const int* in_sizes, int n_in,
                              void* d_out, int out_size, void* d_ws, size_t ws_size,
                              hipStream_t stream) {
    const float* x     = (const float*)d_in[0];
    const float* alpha = (const float*)d_in[1];
    const int*   ei[3] = {(const int*)d_in[2], (const int*)d_in[3], (const int*)d_in[4]};
    const int    E     = in_sizes[2] / 2;   // 800000
    float* out = (float*)d_out;

    const size_t NF = (size_t)NUM_NODES * DIM;     // 3.2M floats per buffer
    float* ws    = (float*)d_ws;
    float* xnorm = ws;                              // 12.8 MB
    float* t0    = xnorm + NF;                      // 12.8 MB
    float* t1    = t0 + NF;                         // 12.8 MB
    float* dinv  = t1 + NF;                         // 200 KB (deg, then d^-1/2)
    float* w3    = dinv + NUM_NODES;                // 3 floats

    const int zb_big   = (int)((NF / 4 + 255) / 256);            // zero 3.2M floats
    const int zb_deg   = (NUM_NODES / 4 + 255) / 256;
    const int nb_norm  = (TILES * 32 + 255) / 256;               // 1 wave per 16-node tile
    const int nb_deg   = (E + 255) / 256;
    const int nb_dinv  = (NUM_NODES + 255) / 256;
    const int nb_conv  = (int)(((size_t)E * 16 + 255) / 256);    // 16 lanes per edge

    k_weights<<<1, 64, 0, stream>>>(alpha, w3);
    k_norm<<<nb_norm, 256, 0, stream>>>(x, xnorm, w3, -1);       // x = l2_normalize(x)
    k_zero<<<zb_big, 256, 0, stream>>>(out, (int)(NF / 4));

    for (int e = 0; e < 3; ++e) {
        // per-edge-set degree -> d^-1/2 (same for both layers)
        k_zero<<<zb_deg, 256, 0, stream>>>(dinv, NUM_NODES / 4);
        k_deg<<<nb_deg, 256, 0, stream>>>(ei[e], dinv, E);
        k_dinv<<<nb_dinv, 256, 0, stream>>>(dinv, NUM_NODES);

        // layer 1: xnorm -> t0
        k_zero<<<zb_big, 256, 0, stream>>>(t0, (int)(NF / 4));
        k_lgconv<<<nb_conv, 256, 0, stream>>>(ei[e], dinv, xnorm, t0, E);
        // layer 2: t0 -> t1
        k_zero<<<zb_big, 256, 0, stream>>>(t1, (int)(NF / 4));
        k_lgconv<<<nb_conv, 256, 0, stream>>>(ei[e], dinv, t0, t1, E);

        // out += w[e] * l2_normalize(t1)
        k_norm<<<nb_norm, 256, 0, stream>>>(t1, out, w3, e);
    }
}